// ScaledDotProductAttention_81260781240908
// MI455X (gfx1250) — compile-verified
//
#include <hip/hip_runtime.h>
#include <hip/hip_bf16.h>
#include <math.h>

typedef __attribute__((ext_vector_type(2))) float v2f;
typedef __attribute__((ext_vector_type(4))) float v4f;
typedef __attribute__((ext_vector_type(8))) float v8f;

#define S_LEN 2048
#define DK    64
#define TILE_M 16

// One wave32 per 16-row tile of one (b,h). LDS holds the full 16 x 2048 fp32
// score strip (128 KB of the 320 KB/WGP budget) so softmax is single-pass.
__global__ __launch_bounds__(32) void
attn_fwd_wmma(const float* __restrict__ q,
              const float* __restrict__ k,   // [B,H,DK,S] (pre-transposed)
              const float* __restrict__ v,   // [B,H,S,DK]
              float* __restrict__ ctx_out,   // [B,H,S,DK]
              float* __restrict__ attn_out)  // [B,H,S,S]
{
    extern __shared__ float smem[];
    float* sc   = smem;                    // [16][2048] scores -> probabilities
    float* red  = smem + TILE_M * S_LEN;   // 32 partial maxes
    float* red2 = red + 32;                // 32 partial sums
    float* mArr = red2 + 32;               // 16 row maxes
    float* lArr = mArr + 16;               // 16 row sums

    const int lane = threadIdx.x;
    const int lo = lane & 15;   // column / row-in-half selector
    const int hi = lane >> 4;   // half-wave selector

    const int bh   = blockIdx.x >> 7;   // 0..31  (b*16 + h)
    const int tile = blockIdx.x & 127;  // 0..127
    const int row0 = tile * TILE_M;

    const float* qb = q + ((size_t)bh * S_LEN + row0) * DK;
    const float* kb = k + (size_t)bh * DK * S_LEN;           // kb[d*S + t]
    const float* vb = v + (size_t)bh * S_LEN * DK;           // vb[t*DK + d]
    float* ctx  = ctx_out  + ((size_t)bh * S_LEN + row0) * DK;
    float* attn = attn_out + ((size_t)bh * S_LEN + row0) * S_LEN;

    const float scale = 0.125f;  // 1/sqrt(64), folded into Q fragments

    // ---- Preload Q as 16 A-fragments (K=4 slices), pre-scaled -------------
    // A 16x4 fp32 layout: lanes 0-15 row=lo hold K={0,1}; lanes 16-31 K={2,3}.
    v2f qa[16];
    #pragma unroll
    for (int dd = 0; dd < 16; ++dd) {
        const float* p = qb + lo * DK + dd * 4 + hi * 2;
        v2f a; a.x = p[0] * scale; a.y = p[1] * scale;
        qa[dd] = a;
    }

    // ---- Phase 1: scores = (Q*scale) @ K  ->  LDS -------------------------
    #pragma unroll 1
    for (int t0 = 0; t0 < S_LEN; t0 += 16) {
        v8f c = {};
        #pragma unroll
        for (int dd = 0; dd < 16; ++dd) {
            const int d0 = dd * 4 + hi * 2;    // B 4x16: rows K, cols N=lo
            v2f b;
            b.x = kb[(size_t)(d0    ) * S_LEN + t0 + lo];
            b.y = kb[(size_t)(d0 + 1) * S_LEN + t0 + lo];
            c = __builtin_amdgcn_wmma_f32_16x16x4_f32(
                    false, qa[dd], false, b, (short)0, c, false, false);
        }
        #pragma unroll
        for (int i = 0; i < 8; ++i) {
            const int row = i + hi * 8;        // C/D layout: VGPR i -> row
            sc[row * S_LEN + t0 + lo] = c[i];
        }
    }
    __syncthreads();

    // ---- Phase 2: row max and sum(exp) (2 lanes per row, b128 scans) ------
    {
        const int r = lane >> 1;
        const int h = lane & 1;
        const v4f* rowp = reinterpret_cast<const v4f*>(sc + r * S_LEN + h * 1024);
        float m = -INFINITY;
        for (int c = 0; c < 256; ++c) {
            v4f t = rowp[c];
            m = fmaxf(m, fmaxf(fmaxf(t.x, t.y), fmaxf(t.z, t.w)));
        }
        red[lane] = m;
        __syncthreads();
        const float mm = fmaxf(red[2 * r], red[2 * r + 1]);
        float s = 0.0f;
        for (int c = 0; c < 256; ++c) {
            v4f t = rowp[c];
            s += __expf(t.x - mm) + __expf(t.y - mm)
               + __expf(t.z - mm) + __expf(t.w - mm);
        }
        red2[lane] = s;
        __syncthreads();
        if (h == 0) {
            mArr[r] = mm;
            lArr[r] = red2[2 * r] + red2[2 * r + 1];
        }
        __syncthreads();
    }

    // ---- Phase 3: normalize; stream attn out as NT b128; keep P in LDS ----
    // attn is 537 MB with zero reuse: non-temporal stores keep it from
    // rinsing K/V (which are re-read ~128x per head) out of the 192 MB L2.
    #pragma unroll 1
    for (int r = 0; r < 16; ++r) {
        const float mm  = mArr[r];
        const float inv = 1.0f / lArr[r];
        float* srow = sc + r * S_LEN;
        float* arow = attn + (size_t)r * S_LEN;
        #pragma unroll 1
        for (int cb = 0; cb < S_LEN; cb += 128) {
            const int c = cb + lane * 4;
            v4f t = *reinterpret_cast<const v4f*>(srow + c);
            v4f p;
            p.x = __expf(t.x - mm) * inv;
            p.y = __expf(t.y - mm) * inv;
            p.z = __expf(t.z - mm) * inv;
            p.w = __expf(t.w - mm) * inv;
            *reinterpret_cast<v4f*>(srow + c) = p;
            __builtin_nontemporal_store(p, reinterpret_cast<v4f*>(arow + c));
        }
    }
    __syncthreads();

    // ---- Phase 4: context = P @ V  (M=16, N=64 via 4 tiles, K=2048) -------
    v8f acc[4] = {};
    #pragma unroll 1
    for (int step = 0; step < S_LEN / 4; ++step) {
        const int kk = step * 4 + hi * 2;
        v2f a;                                    // A from LDS (P)
        a.x = sc[lo * S_LEN + kk];
        a.y = sc[lo * S_LEN + kk + 1];
        #pragma unroll
        for (int n = 0; n < 4; ++n) {
            v2f b;                                // B 4x16 from V
            b.x = vb[(size_t)(kk    ) * DK + n * 16 + lo];
            b.y = vb[(size_t)(kk + 1) * DK + n * 16 + lo];
            acc[n] = __builtin_amdgcn_wmma_f32_16x16x4_f32(
                         false, a, false, b, (short)0, acc[n], false, false);
        }
    }
    #pragma unroll
    for (int n = 0; n < 4; ++n) {
        #pragma unroll
        for (int i = 0; i < 8; ++i) {
            const int row = i + hi * 8;
            ctx[row * DK + n * 16 + lo] = acc[n][i];
        }
    }
}

extern "C" void kernel_launch(void* const* d_in, const int* in_sizes, int n_in,
                              void* d_out, int out_size, void* d_ws, size_t ws_size,
                              hipStream_t stream) {
    (void)in_sizes; (void)n_in; (void)d_ws; (void)ws_size; (void)out_size;
    const float* q = (const float*)d_in[0];
    const float* k = (const float*)d_in[1];
    const float* v = (const float*)d_in[2];

    const int B = 2, H = 16;
    float* ctx  = (float*)d_out;                                   // [B,H,S,DK]
    float* attn = ctx + (size_t)B * H * S_LEN * DK;                // [B,H,S,S]

    const dim3 grid(B * H * (S_LEN / TILE_M));                     // 4096
    const dim3 block(32);
    const size_t shmem = (size_t)(TILE_M * S_LEN + 96) * sizeof(float);

    attn_fwd_wmma<<<grid, block, shmem, stream>>>(q, k, v, ctx, attn);
}